// Channel_Attention_7610682049054
// MI455X (gfx1250) — compile-verified
//
#include <hip/hip_runtime.h>

// Problem constants (match reference)
#define Bb 8
#define Nn 16384
#define Cc 512

typedef __attribute__((ext_vector_type(2))) float v2f;
typedef __attribute__((ext_vector_type(8))) float v8f;

// Only probe the AMDGCN builtin on the DEVICE pass; host pass (x86) has no
// AMDGPU builtins and must not trip this check.
#if defined(__HIP_DEVICE_COMPILE__) && defined(__AMDGCN__)
#if !__has_builtin(__builtin_amdgcn_wmma_f32_16x16x4_f32)
#error "device: __builtin_amdgcn_wmma_f32_16x16x4_f32 not available"
#endif
#endif

__device__ __forceinline__ v8f wmma_k4(v2f a, v2f b, v8f c) {
#if defined(__HIP_DEVICE_COMPILE__)
  // 8 args: (neg_a, A, neg_b, B, c_mod, C, reuse_a, reuse_b)
  return __builtin_amdgcn_wmma_f32_16x16x4_f32(false, a, false, b, (short)0, c,
                                               false, false);
#else
  return c;  // host stub (never executed)
#endif
}

// ---------------------------------------------------------------------------
// Kernel 1: energy[b,i,j] = sum_n x[b,n,i] * x[b,n,j]   (C x C Gram per batch)
// WG = 256 thr (8 waves, 4x2), tile = 128(i) x 64(j); each wave: 2x2 WMMA tiles.
// A/B f32 16x16x4 fragments: lanes 0-15 hold K=0,1 ; lanes 16-31 hold K=2,3.
// ---------------------------------------------------------------------------
__global__ __launch_bounds__(256)
void gram_kernel(const float* __restrict__ x, float* __restrict__ energy) {
  const int b    = blockIdx.z;
  const int i0   = blockIdx.x * 128;
  const int j0   = blockIdx.y * 64;
  const int wave = threadIdx.x >> 5;
  const int lane = threadIdx.x & 31;
  const int l15  = lane & 15;
  const int hi   = lane >> 4;          // 0: K=0,1  1: K=2,3
  const int wr   = wave & 3;           // i direction (4 waves)
  const int wc   = wave >> 2;          // j direction (2 waves)

  const float* xb = x + (size_t)b * Nn * Cc;

  const int ia = i0 + wr * 32 + l15;   // A channels (tile0); tile1 = +16
  const int ja = j0 + wc * 32 + l15;   // B channels

  v8f acc00 = {}, acc01 = {}, acc10 = {}, acc11 = {};

  for (int n = 0; n < Nn; n += 4) {
    const float* r0 = xb + (size_t)(n + 2 * hi) * Cc;  // row n or n+2
    const float* r1 = r0 + Cc;                          // row n+1 or n+3
    v2f a0, a1, b0, b1;
    a0.x = r0[ia];      a0.y = r1[ia];        // coalesced: 16 lanes x 4B = 64B
    a1.x = r0[ia + 16]; a1.y = r1[ia + 16];
    b0.x = r0[ja];      b0.y = r1[ja];
    b1.x = r0[ja + 16]; b1.y = r1[ja + 16];
    acc00 = wmma_k4(a0, b0, acc00);
    acc01 = wmma_k4(a0, b1, acc01);
    acc10 = wmma_k4(a1, b0, acc10);
    acc11 = wmma_k4(a1, b1, acc11);
  }

  // C/D layout: VGPR g -> M = g (lanes 0-15) / g+8 (lanes 16-31), N = lane%16
  float* eb = energy + (size_t)b * Cc * Cc;
  const int ibase = i0 + wr * 32 + hi * 8;
  const int jbase = j0 + wc * 32 + l15;
#pragma unroll
  for (int g = 0; g < 8; ++g) {
    eb[(size_t)(ibase + g) * Cc + (jbase)]           = acc00[g];
    eb[(size_t)(ibase + g) * Cc + (jbase + 16)]      = acc01[g];
    eb[(size_t)(ibase + 16 + g) * Cc + (jbase)]      = acc10[g];
    eb[(size_t)(ibase + 16 + g) * Cc + (jbase + 16)] = acc11[g];
  }
}

// ---------------------------------------------------------------------------
// Kernel 2: attention = softmax(rowmax - energy) == softmax(-energy), row-wise.
// Stable: p_j = exp(min_row(e) - e_j) / sum.  In-place on the workspace.
// One block (256 thr) per (b, i) row of 512 elements.
// ---------------------------------------------------------------------------
__global__ __launch_bounds__(256)
void softmax_kernel(float* __restrict__ e) {
  __shared__ float red[256];
  float* er   = e + (size_t)blockIdx.x * Cc;  // blockIdx.x = b*C + i
  const int t = threadIdx.x;

  float e0 = er[t];
  float e1 = er[t + 256];

  red[t] = fminf(e0, e1);
  __syncthreads();
  for (int s = 128; s > 0; s >>= 1) {
    if (t < s) red[t] = fminf(red[t], red[t + s]);
    __syncthreads();
  }
  const float mn = red[0];
  __syncthreads();

  const float p0 = __expf(mn - e0);
  const float p1 = __expf(mn - e1);
  red[t] = p0 + p1;
  __syncthreads();
  for (int s = 128; s > 0; s >>= 1) {
    if (t < s) red[t] += red[t + s];
    __syncthreads();
  }
  const float inv = 1.0f / red[0];

  er[t]       = p0 * inv;
  er[t + 256] = p1 * inv;
}

// ---------------------------------------------------------------------------
// Kernel 3: out[b,n,i] = gamma * sum_j x[b,n,j]*att[b,i,j] + x[b,n,i]
// WG = 256 thr, output tile 32(n) x 256(i); 8 waves each own a 32-wide i strip
// (2x2 WMMA tiles). x slab (32 x 128 per j-chunk) staged in LDS via
// GLOBAL_LOAD_ASYNC_TO_LDS_B128 (ASYNCcnt, no VGPR round-trip); +4-float pad
// (stride 132 -> bank = 4*row + col: conflict-free b64 fragment reads).
// ---------------------------------------------------------------------------
#define PADC 132

__global__ __launch_bounds__(256)
void apply_kernel(const float* __restrict__ x, const float* __restrict__ att,
                  const float* __restrict__ gptr, float* __restrict__ out) {
  __shared__ float xs[32 * PADC];

  const int b    = blockIdx.z;
  const int n0   = blockIdx.x * 32;
  const int i0   = blockIdx.y * 256;
  const int wave = threadIdx.x >> 5;
  const int lane = threadIdx.x & 31;
  const int l15  = lane & 15;
  const int hi   = lane >> 4;
  const float gamma = gptr[0];

  const float* xb = x + ((size_t)b * Nn + n0) * Cc;
  const float* ab = att + (size_t)b * Cc * Cc;
  const int iw    = i0 + wave * 32;   // this wave's i base

  v8f acc[2][2] = {{{}, {}}, {{}, {}}};

  for (int jc = 0; jc < Cc; jc += 128) {
    __syncthreads();  // previous chunk's consumers are done before overwrite
    // Stage x[n0..n0+32) x [jc..jc+128) -> LDS with async b128 copies:
    // 16B-aligned on both sides (528B LDS row stride, col % 4 == 0).
#pragma unroll
    for (int r = 0; r < 4; ++r) {
      const int idx = ((int)threadIdx.x + 256 * r) * 4;  // flat float idx
      const int row = idx >> 7;                          // /128
      const int col = idx & 127;
      const float* src = &xb[(size_t)row * Cc + jc + col];
      const unsigned dst = (unsigned)(uintptr_t)(&xs[row * PADC + col]);
      asm volatile("global_load_async_to_lds_b128 %0, %1, off"
                   :
                   : "v"(dst), "v"(src)
                   : "memory");
    }
    asm volatile("s_wait_asynccnt 0" ::: "memory");  // my copies hit LDS
    __syncthreads();                                  // everyone's copies

    for (int kk = 0; kk < 128; kk += 4) {
      // A: x rows (n), K along j  -> b64 from LDS
      const v2f a0 = *(const v2f*)&xs[l15 * PADC + kk + 2 * hi];
      const v2f a1 = *(const v2f*)&xs[(16 + l15) * PADC + kk + 2 * hi];
      // B: col = output channel i, K along j -> per-lane b64 from global
      // (att is 8 MB total, L2-resident, reused 512x across n tiles)
      const v2f b0 =
          *(const v2f*)&ab[(size_t)(iw + l15) * Cc + jc + kk + 2 * hi];
      const v2f b1 =
          *(const v2f*)&ab[(size_t)(iw + 16 + l15) * Cc + jc + kk + 2 * hi];
      acc[0][0] = wmma_k4(a0, b0, acc[0][0]);
      acc[0][1] = wmma_k4(a0, b1, acc[0][1]);
      acc[1][0] = wmma_k4(a1, b0, acc[1][0]);
      acc[1][1] = wmma_k4(a1, b1, acc[1][1]);
    }
  }

  // Epilogue: out = gamma*acc + x (residual), coalesced 64B per half-wave.
  float* ob = out + ((size_t)b * Nn + n0) * Cc;
#pragma unroll
  for (int tn = 0; tn < 2; ++tn) {
#pragma unroll
    for (int ti = 0; ti < 2; ++ti) {
#pragma unroll
      for (int g = 0; g < 8; ++g) {
        const int rn = tn * 16 + hi * 8 + g;      // n within tile
        const int c  = iw + ti * 16 + l15;        // channel
        const size_t off = (size_t)rn * Cc + c;
        ob[off] = gamma * acc[tn][ti][g] + xb[off];
      }
    }
  }
}

// ---------------------------------------------------------------------------
extern "C" void kernel_launch(void* const* d_in, const int* in_sizes, int n_in,
                              void* d_out, int out_size, void* d_ws,
                              size_t ws_size, hipStream_t stream) {
  (void)in_sizes; (void)n_in; (void)out_size; (void)ws_size;
  const float* x     = (const float*)d_in[0];
  const float* gamma = (const float*)d_in[1];
  float* out         = (float*)d_out;
  float* energy      = (float*)d_ws;  // B*C*C floats = 8 MB, becomes attention

  dim3 g1(Cc / 128, Cc / 64, Bb);          // (4, 8, 8)
  gram_kernel<<<g1, 256, 0, stream>>>(x, energy);

  softmax_kernel<<<Bb * Cc, 256, 0, stream>>>(energy);

  dim3 g3(Nn / 32, Cc / 256, Bb);          // (512, 2, 8)
  apply_kernel<<<g3, 256, 0, stream>>>(x, energy, gamma, out);
}